// GCN_83124797047020
// MI455X (gfx1250) — compile-verified
//
#include <hip/hip_runtime.h>
#include <math.h>

typedef __attribute__((ext_vector_type(2))) float v2f;
typedef __attribute__((ext_vector_type(8))) float v8f;

#define GCN_N_NODES 100000

// ---------------- degree / normalization ----------------

__global__ void gcn_set_deg(float* __restrict__ deg, int n) {
    int i = blockIdx.x * blockDim.x + threadIdx.x;
    if (i < n) deg[i] = 1.0f;  // self-loop contributes 1 to every node's degree
}

__global__ void gcn_deg_edges(const int* __restrict__ dst, float* __restrict__ deg, int E) {
    int e = blockIdx.x * blockDim.x + threadIdx.x;
    if (e < E) atomicAdd(&deg[dst[e]], 1.0f);
}

__global__ void gcn_dinv(const float* __restrict__ deg, float* __restrict__ dinv, int n) {
    int i = blockIdx.x * blockDim.x + threadIdx.x;
    if (i < n) {
        float d = deg[i];
        dinv[i] = (d > 0.0f) ? rsqrtf(d) : 0.0f;
    }
}

__global__ void gcn_edge_norm(const int* __restrict__ src, const int* __restrict__ dst,
                              const float* __restrict__ dinv, float* __restrict__ enorm,
                              int E) {
    int e = blockIdx.x * blockDim.x + threadIdx.x;
    if (e < E) enorm[e] = dinv[src[e]] * dinv[dst[e]];
}

// ---------------- dense transform: C[M,N] = A[M,K] * B[K,N] via V_WMMA_F32_16X16X4_F32 ----
// Block = 128 threads = 4 waves tiling along M; ntile = blockIdx.y is block-uniform, so
// the 16-column B panel is staged once into LDS (transposed, stride K+4 -> conflict-free
// b64 reads across all 64 banks). A is streamed with non-temporal b64 loads.
// All guards are wave-uniform so EXEC is all-ones at each WMMA (ISA requirement).

template <int RELU, int KT>
__global__ void gcn_gemm_wmma_f32(const float* __restrict__ A,
                                  const float* __restrict__ B,
                                  float* __restrict__ C,
                                  int Mtiles, int N) {
    constexpr int S = KT + 4;            // LDS row stride: S % 64 == 4 for K=256/64
    __shared__ float Bs[16 * S];         // Bs[n][k] (transposed panel)

    const int wave  = threadIdx.x >> 5;
    const int lane  = threadIdx.x & 31;
    const int mtile = blockIdx.x * (blockDim.x >> 5) + wave;
    const int ntile = blockIdx.y;

    // cooperative load of B panel, transposed into LDS (float4 along N)
    for (int idx = threadIdx.x; idx < KT * 4; idx += blockDim.x) {
        int k  = idx >> 2;
        int n4 = (idx & 3) << 2;
        float4 v = *(const float4*)(B + (size_t)k * N + ntile * 16 + n4);
        Bs[(n4 + 0) * S + k] = v.x;
        Bs[(n4 + 1) * S + k] = v.y;
        Bs[(n4 + 2) * S + k] = v.z;
        Bs[(n4 + 3) * S + k] = v.w;
    }
    __syncthreads();

    if (mtile < Mtiles) {                // wave-uniform guard (after the barrier)
        const int m  = lane & 15;        // A row within tile
        const int kb = (lane >> 4) << 1; // lanes 0-15: K pair {0,1}; lanes 16-31: {2,3}
        const int n  = lane & 15;        // B col within tile

        const float* __restrict__ arow = A + (size_t)(mtile * 16 + m) * KT + kb;
        const float* __restrict__ brow = Bs + n * S + kb;

        v8f c = {0.f, 0.f, 0.f, 0.f, 0.f, 0.f, 0.f, 0.f};

#pragma unroll 8
        for (int k = 0; k < KT; k += 4) {
            v2f a = __builtin_nontemporal_load((const v2f*)(arow + k));
            if (RELU) {
                a.x = fmaxf(a.x, 0.0f);
                a.y = fmaxf(a.y, 0.0f);
            }
            v2f b = *(const v2f*)(brow + k);          // Bs[n][k+kb], Bs[n][k+kb+1]
            c = __builtin_amdgcn_wmma_f32_16x16x4_f32(
                    /*neg_a=*/false, a, /*neg_b=*/false, b,
                    /*c_mod=*/(short)0, c, /*reuse_a=*/false, /*reuse_b=*/false);
        }

        // C/D layout: VGPR r -> row r (lanes 0-15) / row r+8 (lanes 16-31), col = lane&15
        float* __restrict__ crow =
            C + (size_t)(mtile * 16 + ((lane >> 4) << 3)) * N + ntile * 16 + (lane & 15);
#pragma unroll
        for (int r = 0; r < 8; ++r) crow[(size_t)r * N] = c[r];
    }
}

// ---------------- self-loop + bias init: out = dinv[i]^2 * xl + bias ----------------

template <int F>
__global__ void gcn_init_out(const float* __restrict__ xl,
                             const float* __restrict__ dinv,
                             const float* __restrict__ bias,
                             float* __restrict__ out, int n) {
    constexpr int Q = F / 4;
    int i = blockIdx.x * blockDim.x + threadIdx.x;  // over n*Q float4 chunks
    if (i >= n * Q) return;
    int node = i / Q;
    int f    = (i % Q) * 4;
    float di = dinv[node];
    float s  = di * di;
    float4 v = *(const float4*)(xl + (size_t)i * 4);
    float4 b = *(const float4*)(bias + f);
    float4 o = make_float4(s * v.x + b.x, s * v.y + b.y, s * v.z + b.z, s * v.w + b.w);
    *(float4*)(out + (size_t)i * 4) = o;
}

// ---------------- edge scatter-add: out[dst] += enorm[e] * xl[src] ----------

template <int F>
__global__ void gcn_edge_agg(const int* __restrict__ src,
                             const int* __restrict__ dst,
                             const float* __restrict__ enorm,
                             const float* __restrict__ xl,
                             float* __restrict__ out, int E) {
    constexpr int TPE = F / 4;  // threads per edge, 4 features each
    long long gid = (long long)blockIdx.x * blockDim.x + threadIdx.x;
    int e = (int)(gid / TPE);
    if (e >= E) return;
    int f = (int)(gid % TPE) * 4;

    int s = src[e];
    int d = dst[e];
    float nm = enorm[e];

    float4 v = *(const float4*)(xl + (size_t)s * F + f);
    float* o = out + (size_t)d * F + f;
    atomicAdd(o + 0, nm * v.x);
    atomicAdd(o + 1, nm * v.y);
    atomicAdd(o + 2, nm * v.z);
    atomicAdd(o + 3, nm * v.w);
}

// ---------------- log_softmax over 16 classes, one thread per node ----------------

__global__ void gcn_log_softmax16(const float* __restrict__ h, float* __restrict__ out, int n) {
    int i = blockIdx.x * blockDim.x + threadIdx.x;
    if (i >= n) return;
    const float4* p = (const float4*)(h + (size_t)i * 16);
    float4 r0 = p[0], r1 = p[1], r2 = p[2], r3 = p[3];
    float v[16] = {r0.x, r0.y, r0.z, r0.w, r1.x, r1.y, r1.z, r1.w,
                   r2.x, r2.y, r2.z, r2.w, r3.x, r3.y, r3.z, r3.w};
    float mx = v[0];
#pragma unroll
    for (int j = 1; j < 16; ++j) mx = fmaxf(mx, v[j]);
    float ssum = 0.0f;
#pragma unroll
    for (int j = 0; j < 16; ++j) ssum += expf(v[j] - mx);
    float lse = mx + logf(ssum);
    float4* q = (float4*)(out + (size_t)i * 16);
    q[0] = make_float4(v[0] - lse, v[1] - lse, v[2] - lse, v[3] - lse);
    q[1] = make_float4(v[4] - lse, v[5] - lse, v[6] - lse, v[7] - lse);
    q[2] = make_float4(v[8] - lse, v[9] - lse, v[10] - lse, v[11] - lse);
    q[3] = make_float4(v[12] - lse, v[13] - lse, v[14] - lse, v[15] - lse);
}

// ---------------- launch ----------------

extern "C" void kernel_launch(void* const* d_in, const int* in_sizes, int n_in,
                              void* d_out, int out_size, void* d_ws, size_t ws_size,
                              hipStream_t stream) {
    const float* x  = (const float*)d_in[0];   // [N, 256]
    const int*   ei = (const int*)d_in[1];     // [2, E]
    const float* W1 = (const float*)d_in[2];   // [256, 64]
    const float* b1 = (const float*)d_in[3];   // [64]
    const float* W2 = (const float*)d_in[4];   // [64, 16]
    const float* b2 = (const float*)d_in[5];   // [16]
    float* out = (float*)d_out;                // [N, 16] log-softmax

    const int Nn = GCN_N_NODES;
    const int E  = in_sizes[1] / 2;
    const int* src = ei;
    const int* dst = ei + E;

    // workspace layout (floats)
    float* ws    = (float*)d_ws;
    float* deg   = ws;                           // Nn
    float* dinv  = deg + Nn;                     // Nn
    float* enorm = dinv + Nn;                    // E
    float* xl1   = enorm + (size_t)E;            // Nn*64
    float* h1    = xl1 + (size_t)Nn * 64;        // Nn*64
    float* xl2   = h1 + (size_t)Nn * 64;         // Nn*16
    float* h2    = xl2 + (size_t)Nn * 16;        // Nn*16

    // symmetric normalization coefficients
    gcn_set_deg<<<(Nn + 255) / 256, 256, 0, stream>>>(deg, Nn);
    gcn_deg_edges<<<(E + 255) / 256, 256, 0, stream>>>(dst, deg, E);
    gcn_dinv<<<(Nn + 255) / 256, 256, 0, stream>>>(deg, dinv, Nn);
    gcn_edge_norm<<<(E + 255) / 256, 256, 0, stream>>>(src, dst, dinv, enorm, E);

    const int Mtiles = Nn / 16;                  // 6250 (exact)
    const int mblocks = (Mtiles + 3) / 4;        // 4 waves per block along M

    // layer 1: xl1 = x @ W1 (WMMA), h1 = dinv^2*xl1 + b1, then edge scatter-add
    gcn_gemm_wmma_f32<0, 256><<<dim3(mblocks, 64 / 16), 128, 0, stream>>>(
        x, W1, xl1, Mtiles, 64);
    gcn_init_out<64><<<(Nn * 16 + 255) / 256, 256, 0, stream>>>(xl1, dinv, b1, h1, Nn);
    {
        long long tot = (long long)E * 16;  // 16 threads per edge
        gcn_edge_agg<64><<<(unsigned)((tot + 255) / 256), 256, 0, stream>>>(
            src, dst, enorm, xl1, h1, E);
    }

    // layer 2: xl2 = relu(h1) @ W2 (ReLU fused into A-load), h2 init, edge scatter-add
    gcn_gemm_wmma_f32<1, 64><<<dim3(mblocks, 16 / 16), 128, 0, stream>>>(
        h1, W2, xl2, Mtiles, 16);
    gcn_init_out<16><<<(Nn * 4 + 255) / 256, 256, 0, stream>>>(xl2, dinv, b2, h2, Nn);
    {
        long long tot = (long long)E * 4;   // 4 threads per edge
        gcn_edge_agg<16><<<(unsigned)((tot + 255) / 256), 256, 0, stream>>>(
            src, dst, enorm, xl2, h2, E);
    }

    gcn_log_softmax16<<<(Nn + 255) / 256, 256, 0, stream>>>(h2, out, Nn);
}